// ThermalDeltaModule_38036230374023
// MI455X (gfx1250) — compile-verified
//
#include <hip/hip_runtime.h>
#include <math.h>

// Problem constants (fixed by the reference).
constexpr int B    = 64;
constexpr int T    = 2048;
constexpr int C    = 512;
constexpr int NSEG = 16;
constexpr int TSEG = T / NSEG;   // 128
constexpr int K1   = 3 * C;      // 1536
constexpr int N1   = C;          // 512
constexpr int K2   = C;          // 512
constexpr int N2   = C / 2;      // 256

typedef float v2f __attribute__((ext_vector_type(2)));
typedef float v8f __attribute__((ext_vector_type(8)));

// ---------------------------------------------------------------------------
// Kernel 1: segmented ragged pooling partials (bandwidth-bound, 256 MiB read).
// grid (NSEG, B), block 128 threads; each thread owns 4 consecutive channels
// -> float4 (GLOBAL_LOAD_B128) fully coalesced: 2 KB contiguous per t-row.
// ---------------------------------------------------------------------------
__global__ void pool_partial_kernel(const float* __restrict__ x,
                                    const int* __restrict__ lengths,
                                    float* __restrict__ partial) {
  const int seg = blockIdx.x;
  const int b   = blockIdx.y;
  const int c4  = threadIdx.x * 4;

  const int L = lengths[b];
  const int q = L / 4;
  const int p = (3 * L) / 4;

  const int t0 = seg * TSEG;
  int t1 = t0 + TSEG;
  if (t1 > L) t1 = L;  // t >= L never contributes to any mask

  float sb[4] = {0.f, 0.f, 0.f, 0.f};
  float sf[4] = {0.f, 0.f, 0.f, 0.f};
  float mx[4];
  const float ninf = -__builtin_inff();
  mx[0] = ninf; mx[1] = ninf; mx[2] = ninf; mx[3] = ninf;

  const float* xp = x + ((size_t)b * T + t0) * C + c4;
  for (int t = t0; t < t1; ++t, xp += C) {
    const float4 v = *(const float4*)xp;
    const float vv[4] = {v.x, v.y, v.z, v.w};
    if (t < q) {
      #pragma unroll
      for (int i = 0; i < 4; ++i) sb[i] += vv[i];
    } else if (t < p) {
      #pragma unroll
      for (int i = 0; i < 4; ++i) mx[i] = fmaxf(mx[i], vv[i]);
    } else {
      #pragma unroll
      for (int i = 0; i < 4; ++i) sf[i] += vv[i];
    }
  }

  float* base = partial + (((size_t)b * NSEG + seg) * 3) * C + c4;
  *(float4*)(base + 0 * C) = make_float4(sb[0], sb[1], sb[2], sb[3]);
  *(float4*)(base + 1 * C) = make_float4(mx[0], mx[1], mx[2], mx[3]);
  *(float4*)(base + 2 * C) = make_float4(sf[0], sf[1], sf[2], sf[3]);
}

// ---------------------------------------------------------------------------
// Kernel 2: reduce NSEG partials per (b,c), apply 1/q and 1/(L-p), emit feat
// laid out as [baseline(C) | peak(C) | final(C)] per batch row (B x 3C).
// grid (B), block C.
// ---------------------------------------------------------------------------
__global__ void pool_reduce_kernel(const float* __restrict__ partial,
                                   const int* __restrict__ lengths,
                                   float* __restrict__ feat) {
  const int b = blockIdx.x;
  const int c = threadIdx.x;

  const int L = lengths[b];
  const float qf = (float)(L / 4);
  const float ff = (float)(L - (3 * L) / 4);

  float sb = 0.f, sf = 0.f, mx = -__builtin_inff();
  const float* base = partial + (size_t)b * NSEG * 3 * C + c;
  #pragma unroll
  for (int s = 0; s < NSEG; ++s) {
    const float* ps = base + (size_t)s * 3 * C;
    sb += ps[0];
    mx = fmaxf(mx, ps[C]);
    sf += ps[2 * C];
  }
  float* fb = feat + (size_t)b * K1;
  fb[c]         = sb / qf;
  fb[C + c]     = mx;
  fb[2 * C + c] = sf / ff;
}

// ---------------------------------------------------------------------------
// WMMA GEMM: Out(MxN) = act(A(MxK) @ W(KxN) + bias), full f32 precision via
// V_WMMA_F32_16X16X4_F32. One wave (32 threads) per 16x16 output tile,
// K-loop in steps of 4. EXEC is all-1s (blockDim == 32, no divergence).
//
// A operand layout (ISA 7.12.2, 32-bit A 16x4): lanes 0-15 -> M=lane, K={0,1}
// in v0/v1; lanes 16-31 -> M=lane-16, K={2,3}. B operand symmetric on N.
// D layout: reg r, lane L -> M = r + 8*(L>=16), N = L&15.
// ---------------------------------------------------------------------------
__device__ __forceinline__ v8f wmma_f32_k4(v2f a, v2f b, v8f c) {
  return __builtin_amdgcn_wmma_f32_16x16x4_f32(
      /*neg_a=*/false, a, /*neg_b=*/false, b,
      /*c_mod=*/(short)0, c, /*reuse_a=*/false, /*reuse_b=*/false);
}

__global__ void gemm_bias_act_kernel(const float* __restrict__ A,   // M x K
                                     const float* __restrict__ W,   // K x N
                                     const float* __restrict__ bias,// N
                                     float* __restrict__ Out,       // M x N
                                     int N, int K, int do_relu) {
  const int lane = threadIdx.x;          // 0..31
  const int l16  = lane & 15;
  const int kh   = (lane >> 4) * 2;      // 0 or 2: K sub-pair this half-wave owns

  const int arow = (blockIdx.y << 4) + l16;  // A row this lane feeds
  const int ncol = (blockIdx.x << 4) + l16;  // output/B column this lane feeds

  v8f acc = {};
  const float* Arow = A + (size_t)arow * K;

  for (int k = 0; k < K; k += 4) {
    const int kk = k + kh;                       // even -> 8B-aligned pair
    const v2f a = *(const v2f*)(Arow + kk);      // A[arow][kk], A[arow][kk+1]
    v2f bm;
    bm.x = W[(size_t)kk * N + ncol];             // B[kk][ncol]
    bm.y = W[(size_t)(kk + 1) * N + ncol];       // B[kk+1][ncol]
    acc = wmma_f32_k4(a, bm, acc);
  }

  const float bv = bias[ncol];
  const int rbase = (blockIdx.y << 4) + ((lane >> 4) << 3);  // M of acc[0]
  #pragma unroll
  for (int r = 0; r < 8; ++r) {
    float v = acc[r] + bv;
    if (do_relu) v = fmaxf(v, 0.0f);
    Out[(size_t)(rbase + r) * N + ncol] = v;
  }
}

// ---------------------------------------------------------------------------
// Launch: pool partials -> pool reduce -> GEMM1(relu) -> GEMM2.
// ws layout (floats): partial[B*NSEG*3*C] | feat[B*3C] | h[B*C]   (~6.7 MB)
// ---------------------------------------------------------------------------
extern "C" void kernel_launch(void* const* d_in, const int* in_sizes, int n_in,
                              void* d_out, int out_size, void* d_ws, size_t ws_size,
                              hipStream_t stream) {
  const float* x       = (const float*)d_in[0];
  const int*   lengths = (const int*)  d_in[1];
  const float* W1      = (const float*)d_in[2];
  const float* b1      = (const float*)d_in[3];
  const float* W2      = (const float*)d_in[4];
  const float* b2      = (const float*)d_in[5];
  float*       out     = (float*)d_out;

  float* ws      = (float*)d_ws;
  float* partial = ws;                                    // B*NSEG*3*C
  float* feat    = partial + (size_t)B * NSEG * 3 * C;    // B*K1
  float* h       = feat + (size_t)B * K1;                 // B*C

  pool_partial_kernel<<<dim3(NSEG, B), dim3(C / 4), 0, stream>>>(x, lengths, partial);
  pool_reduce_kernel<<<dim3(B), dim3(C), 0, stream>>>(partial, lengths, feat);

  // h = relu(feat @ W1 + b1): M=64, N=512, K=1536 -> grid (32, 4), 1 wave/tile
  gemm_bias_act_kernel<<<dim3(N1 / 16, B / 16), dim3(32), 0, stream>>>(
      feat, W1, b1, h, N1, K1, 1);
  // out = h @ W2 + b2: M=64, N=256, K=512 -> grid (16, 4)
  gemm_bias_act_kernel<<<dim3(N2 / 16, B / 16), dim3(32), 0, stream>>>(
      h, W2, b2, out, N2, K2, 0);
}